// PNP_12455405159087
// MI455X (gfx1250) — compile-verified
//
#include <hip/hip_runtime.h>
#include <hip/hip_bf16.h>
#include <math.h>
#include <stdint.h>

typedef __attribute__((ext_vector_type(2))) float v2f;
typedef __attribute__((ext_vector_type(8))) float v8f;
typedef __attribute__((ext_vector_type(4))) unsigned int v4u;
typedef __attribute__((ext_vector_type(8))) int v8i;
typedef __attribute__((ext_vector_type(4))) int v4i;

#define BN_TOK 32768      // B*N = 32*1024
#define DIMD   768
#define CTOT   201
#define KPROT  5
#define CK     1005       // 201*5
#define CKPAD  1024
#define NCLS   200

// ---------------- row inverse L2 norm (one block per row) ----------------
__global__ __launch_bounds__(256) void row_inv_norm_k(const float* __restrict__ x,
                                                      float* __restrict__ inv) {
  __shared__ float red[256];
  int row = blockIdx.x;
  int tid = threadIdx.x;
  float s = 0.f;
  for (int i = tid; i < DIMD; i += 256) {
    float v = x[(size_t)row * DIMD + i];
    s += v * v;
  }
  red[tid] = s; __syncthreads();
  for (int off = 128; off > 0; off >>= 1) {
    if (tid < off) red[tid] += red[tid + off];
    __syncthreads();
  }
  if (tid == 0) inv[row] = 1.0f / fmaxf(sqrtf(red[0]), 1e-12f);
}

// ---------------- normalize prototypes into padded ws buffer ----------------
__global__ __launch_bounds__(256) void proto_norm_k(const float* __restrict__ proto,
                                                    float* __restrict__ protoN) {
  __shared__ float red[256];
  __shared__ float invs;
  int row = blockIdx.x;          // 0..1004
  int tid = threadIdx.x;
  float s = 0.f;
  for (int i = tid; i < DIMD; i += 256) {
    float v = proto[(size_t)row * DIMD + i];
    s += v * v;
  }
  red[tid] = s; __syncthreads();
  for (int off = 128; off > 0; off >>= 1) {
    if (tid < off) red[tid] += red[tid + off];
    __syncthreads();
  }
  if (tid == 0) invs = 1.0f / fmaxf(sqrtf(red[0]), 1e-12f);
  __syncthreads();
  float inv = invs;
  for (int i = tid; i < DIMD; i += 256)
    protoN[(size_t)row * DIMD + i] = proto[(size_t)row * DIMD + i] * inv;
}

// ---------------- WMMA GEMM with TDM double-buffered LDS staging ----------------
// logits[32768,1005] = (pt * rowinv) @ protoN^T ; row-scale applied at epilogue.
// Block tile 128(M) x 64(N), Kc=32. 8 waves, each 32x32 via 2x2 WMMA f32 accums.
// TDM pads LDS rows: 32 dwords data + 4 dwords pad -> row stride 36 floats.
#define TM 128
#define TN 64
#define KC 32
#define LDA 36                      // padded LDS row stride in floats (TDM pad)
#define A_TILE_BYTES (TM * LDA * 4) // 18432
#define B_TILE_BYTES (TN * LDA * 4) // 9216
#define GEMM_LDS_BYTES (2 * A_TILE_BYTES + 2 * B_TILE_BYTES) // 55296

// Issue a 2D TDM tile load: tile_rows x 32 f32 elements, tensor row stride 768,
// LDS destination padded (pad_interval=32 dwords, pad_amount=4 dwords).
__device__ __forceinline__ void tdm_load_2d(uint64_t gaddr, unsigned int lds_off,
                                            unsigned int rows_total,
                                            unsigned int tile_rows) {
  v4u g0;
  g0[0] = 1u;                                               // count=1, user mode
  g0[1] = lds_off;                                          // lds_addr (bytes)
  g0[2] = (unsigned int)gaddr;                              // global_addr lo32
  g0[3] = (unsigned int)((gaddr >> 32) & 0x01FFFFFFu) | (2u << 30); // hi25 | type=2
  v8i g1;
  g1[0] = (int)((2u << 16)   // data_size = 4B
              | (1u << 20)   // pad_enable
              | (4u << 22)   // pad_interval -> 32 dwords
              | (3u << 25)); // pad_amount   -> 4 dwords
  g1[1] = (int)(((unsigned int)DIMD & 0xffffu) << 16);      // tensor_dim0 lo16
  g1[2] = (int)(((unsigned int)DIMD >> 16) | ((rows_total & 0xffffu) << 16));
  g1[3] = (int)((rows_total >> 16) | (32u << 16));          // tensor_dim1 hi | tile_dim0=32
  g1[4] = (int)(tile_rows & 0xffffu);                       // tile_dim1, tile_dim2=0
  g1[5] = DIMD;                                             // tensor_dim0_stride lo32
  g1[6] = 0;
  g1[7] = 0;
  v4i z4 = {0, 0, 0, 0};
  v8i z8 = {0, 0, 0, 0, 0, 0, 0, 0};
  __builtin_amdgcn_tensor_load_to_lds(g0, g1, z4, z4, z8, 0);
}

__global__ __launch_bounds__(256) void gemm_logits_k(const float* __restrict__ pt,
                                                     const float* __restrict__ ptinv,
                                                     const float* __restrict__ protoN,
                                                     float* __restrict__ logits) {
  extern __shared__ float smem[];   // [A buf0][A buf1][B buf0][B buf1]
  const int tid  = threadIdx.x;
  const int lane = tid & 31;
  const int hs   = lane >> 4;        // 0 -> K {0,1}, 1 -> K {2,3}
  const int lrow = lane & 15;
  const int wave = tid >> 5;
  const int wm   = wave & 3;         // M quadrant (rows 32*wm)
  const int wn   = wave >> 2;        // N half    (cols 32*wn)
  const size_t mBase = (size_t)blockIdx.x * TM;
  const int    nBase = blockIdx.y * TN;
  const int wid_s = __builtin_amdgcn_readfirstlane(wave);  // scalar wave id

  const uint64_t gA = (uint64_t)(uintptr_t)pt +
                      (uint64_t)(mBase * DIMD) * 4ull;
  const uint64_t gB = (uint64_t)(uintptr_t)protoN +
                      (uint64_t)((size_t)nBase * DIMD) * 4ull;

  v8f acc00 = {}, acc01 = {}, acc10 = {}, acc11 = {};

  if (wid_s == 0) {   // elect wave 0 via scalar branch (TDM ignores EXEC)
    tdm_load_2d(gA, 0u, (unsigned)BN_TOK, TM);
    tdm_load_2d(gB, 2u * A_TILE_BYTES, (unsigned)CKPAD, TN);
  }

  const int NCHUNK = DIMD / KC;      // 24
  for (int c = 0; c < NCHUNK; ++c) {
    if (wid_s == 0) {
      if (c + 1 < NCHUNK) {
        uint64_t koff = (uint64_t)(c + 1) * KC * 4ull;
        unsigned d = (unsigned)((c + 1) & 1);
        tdm_load_2d(gA + koff, d * A_TILE_BYTES, (unsigned)BN_TOK, TM);
        tdm_load_2d(gB + koff, 2u * A_TILE_BYTES + d * B_TILE_BYTES,
                    (unsigned)CKPAD, TN);
        __builtin_amdgcn_s_wait_tensorcnt(2);  // chunk c's two loads complete
      } else {
        __builtin_amdgcn_s_wait_tensorcnt(0);
      }
    }
    __syncthreads();                 // publish chunk c's LDS tiles to all waves

    const float* As = smem + (c & 1) * (TM * LDA);
    const float* Bs = smem + 2 * (TM * LDA) + (c & 1) * (TN * LDA);

    #pragma unroll
    for (int kk = 0; kk < KC; kk += 4) {
      v2f a0 = *reinterpret_cast<const v2f*>(&As[(wm * 32 +      lrow) * LDA + kk + 2 * hs]);
      v2f a1 = *reinterpret_cast<const v2f*>(&As[(wm * 32 + 16 + lrow) * LDA + kk + 2 * hs]);
      v2f b0 = *reinterpret_cast<const v2f*>(&Bs[(wn * 32 +      lrow) * LDA + kk + 2 * hs]);
      v2f b1 = *reinterpret_cast<const v2f*>(&Bs[(wn * 32 + 16 + lrow) * LDA + kk + 2 * hs]);
      acc00 = __builtin_amdgcn_wmma_f32_16x16x4_f32(false, a0, false, b0, (short)0, acc00, false, false);
      acc01 = __builtin_amdgcn_wmma_f32_16x16x4_f32(false, a0, false, b1, (short)0, acc01, false, false);
      acc10 = __builtin_amdgcn_wmma_f32_16x16x4_f32(false, a1, false, b0, (short)0, acc10, false, false);
      acc11 = __builtin_amdgcn_wmma_f32_16x16x4_f32(false, a1, false, b1, (short)0, acc11, false, false);
    }
    __syncthreads();                 // all waves done reading buf (c&1) before overwrite
  }

  // epilogue: D layout -> lanes 0-15: N=lane, M=r ; lanes 16-31: N=lane-16, M=r+8.
  // Apply the per-row 1/||pt_row|| scale here (WMMA is linear in A).
  #pragma unroll
  for (int mi = 0; mi < 2; ++mi) {
    v8f am0 = (mi == 0) ? acc00 : acc10;
    v8f am1 = (mi == 0) ? acc01 : acc11;
    #pragma unroll
    for (int ni = 0; ni < 2; ++ni) {
      v8f a = (ni == 0) ? am0 : am1;
      int col = nBase + wn * 32 + ni * 16 + lrow;
      if (col < CK) {
        #pragma unroll
        for (int r = 0; r < 8; ++r) {
          size_t row = mBase + wm * 32 + mi * 16 + r + 8 * hs;
          logits[row * CK + col] = a[r] * ptinv[row];
        }
      }
    }
  }
}

// ---------------- image_prototype_logits: max over N per (b, c*K+k) ----------------
__global__ __launch_bounds__(256) void ipl_max_k(const float* __restrict__ logits,
                                                 float* __restrict__ ipl) {
  int j = blockIdx.x * 256 + threadIdx.x;
  int b = blockIdx.y;
  if (j >= CK) return;
  const float* base = logits + (size_t)b * 1024 * CK + j;
  float mx = -3.4e38f;
  for (int n = 0; n < 1024; ++n) mx = fmaxf(mx, base[(size_t)n * CK]);
  ipl[(size_t)b * CK + j] = mx;
}

// ---------------- class_logits ----------------
__global__ __launch_bounds__(256) void class_logits_k(const float* __restrict__ sa_w,
                                                      const float* __restrict__ ipl,
                                                      float* __restrict__ out) {
  int i = blockIdx.x * 256 + threadIdx.x;
  if (i >= 32 * NCLS) return;
  int b = i / NCLS, c = i % NCLS;
  float w[KPROT];
  float mx = -3.4e38f;
  #pragma unroll
  for (int k = 0; k < KPROT; ++k) { w[k] = sa_w[c * KPROT + k]; mx = fmaxf(mx, w[k]); }
  float s = 0.f;
  #pragma unroll
  for (int k = 0; k < KPROT; ++k) { w[k] = __expf(w[k] - mx); s += w[k]; }
  float acc = 0.f;
  #pragma unroll
  for (int k = 0; k < KPROT; ++k)
    acc += ipl[(size_t)b * CK + c * KPROT + k] * (w[k] / s * (float)KPROT);
  out[(size_t)b * NCLS + c] = acc / 0.2f;
}

// ---------------- per-token: pseudo label, softmax A, threshold, argmax ----------------
__global__ __launch_bounds__(256) void token_k(const int* __restrict__ sam,
                                               const int* __restrict__ labels,
                                               const float* __restrict__ logits,
                                               float* __restrict__ pseudo_out,
                                               float* __restrict__ partmap_out,
                                               float* __restrict__ Aws,
                                               int* __restrict__ counts) {
  int n = blockIdx.x * 256 + threadIdx.x;
  if (n >= BN_TOK) return;
  int b = n >> 10;
  int c = (sam[n] == 1) ? labels[b] : NCLS;
  pseudo_out[n] = (float)c;
  const float* L = logits + (size_t)n * CK + c * KPROT;
  float l[KPROT];
  float mx = -3.4e38f;
  #pragma unroll
  for (int k = 0; k < KPROT; ++k) { l[k] = L[k]; mx = fmaxf(mx, l[k]); }
  float s = 0.f;
  #pragma unroll
  for (int k = 0; k < KPROT; ++k) { l[k] = __expf(l[k] - mx); s += l[k]; }
  float A[KPROT];
  float amax = -1.f; int aidx = 0;
  #pragma unroll
  for (int k = 0; k < KPROT; ++k) {
    A[k] = l[k] / s;
    if (A[k] > amax) { amax = A[k]; aidx = k; }   // first-max tie-break like argmax
  }
  if (amax < 0.8f) {
    #pragma unroll
    for (int k = 0; k < KPROT; ++k) A[k] = 0.f;
  }
  float s2 = 0.f;
  #pragma unroll
  for (int k = 0; k < KPROT; ++k) s2 += A[k];
  float invs = 1.0f / fmaxf(s2, 1e-12f);
  #pragma unroll
  for (int k = 0; k < KPROT; ++k) Aws[(size_t)n * KPROT + k] = A[k] * invs;
  partmap_out[n] = (float)(aidx + c * KPROT);
  atomicAdd(&counts[c], 1);
}

// ---------------- per-image segment accumulation: acc[c,k,d] += A[n,k]*rawN[n,d] ----------------
__global__ __launch_bounds__(768) void accum_k(const float* __restrict__ raw,
                                               const float* __restrict__ rawinv,
                                               const float* __restrict__ Aws,
                                               const float* __restrict__ pseudo,
                                               const int* __restrict__ labels,
                                               float* __restrict__ accbuf) {
  __shared__ float sA[128 * KPROT];
  __shared__ float sInv[128];
  __shared__ float sCls[128];
  const int b = blockIdx.x;
  const int d = threadIdx.x;       // 0..767
  float aF[KPROT] = {0.f, 0.f, 0.f, 0.f, 0.f};
  float aB[KPROT] = {0.f, 0.f, 0.f, 0.f, 0.f};
  for (int ch = 0; ch < 8; ++ch) {
    const int base = b * 1024 + ch * 128;
    __syncthreads();
    if (d < 640) {
      sA[d] = Aws[(size_t)base * KPROT + d];
    } else {
      int t = d - 640;
      sInv[t] = rawinv[base + t];
      sCls[t] = pseudo[base + t];
    }
    __syncthreads();
    for (int t = 0; t < 128; ++t) {
      float val = raw[(size_t)(base + t) * DIMD + d] * sInv[t];
      bool fg = (sCls[t] != (float)NCLS);   // uniform across the block
      #pragma unroll
      for (int k = 0; k < KPROT; ++k) {
        float a = sA[t * KPROT + k] * val;
        if (fg) aF[k] += a; else aB[k] += a;
      }
    }
  }
  int lab = labels[b];
  #pragma unroll
  for (int k = 0; k < KPROT; ++k)
    atomicAdd(&accbuf[((size_t)lab * KPROT + k) * DIMD + d], aF[k]);
  #pragma unroll
  for (int k = 0; k < KPROT; ++k)
    atomicAdd(&accbuf[((size_t)NCLS * KPROT + k) * DIMD + d], aB[k]);
}

// ---------------- EMA prototype update ----------------
__global__ __launch_bounds__(256) void ema_k(const float* __restrict__ proto,
                                             const float* __restrict__ accbuf,
                                             const int* __restrict__ counts,
                                             float* __restrict__ out) {
  int i = blockIdx.x * 256 + threadIdx.x;
  if (i >= CTOT * KPROT * DIMD) return;
  int c = i / (KPROT * DIMD);
  float p = proto[i];
  out[i] = (counts[c] > 0) ? (0.999f * p + (1.0f - 0.999f) * accbuf[i]) : p;
}

extern "C" void kernel_launch(void* const* d_in, const int* in_sizes, int n_in,
                              void* d_out, int out_size, void* d_ws, size_t ws_size,
                              hipStream_t stream) {
  const float* patch  = (const float*)d_in[0];   // [32,1024,768]
  const float* raw    = (const float*)d_in[1];   // [32,1024,768]
  const float* proto  = (const float*)d_in[2];   // [201,5,768]
  const float* sa_w   = (const float*)d_in[3];   // [200,5]
  const int*   labels = (const int*)d_in[4];     // [32]
  const int*   sam    = (const int*)d_in[5];     // [32,32,32]

  float* out = (float*)d_out;
  float* cls_out = out;                       // 6400
  float* ipl_out = out + 6400;                // 32160
  float* logits  = out + 38560;               // 32931840
  float* partmap = out + 32970400;            // 32768
  float* pnew    = out + 33003168;            // 771840
  float* pseudo  = out + 33775008;            // 32768

  float* ws = (float*)d_ws;
  float* protoN  = ws;                        // 1024*768 = 786432 (padded, zeroed)
  float* ptinv   = ws + 786432;               // 32768
  float* rawinv  = ws + 819200;               // 32768
  float* Aws     = ws + 851968;               // 32768*5 = 163840
  float* accbuf  = ws + 1015808;              // 771840
  int*   counts  = (int*)(ws + 1787648);      // 201 ints

  (void)hipMemsetAsync(protoN, 0, (size_t)786432 * sizeof(float), stream);
  (void)hipMemsetAsync(accbuf, 0, (size_t)771840 * sizeof(float), stream);
  (void)hipMemsetAsync(counts, 0, (size_t)201 * sizeof(int), stream);

  row_inv_norm_k<<<BN_TOK, 256, 0, stream>>>(patch, ptinv);
  row_inv_norm_k<<<BN_TOK, 256, 0, stream>>>(raw, rawinv);
  proto_norm_k<<<CK, 256, 0, stream>>>(proto, protoN);

  gemm_logits_k<<<dim3(BN_TOK / TM, CKPAD / TN), 256, GEMM_LDS_BYTES, stream>>>(
      patch, ptinv, protoN, logits);

  ipl_max_k<<<dim3((CK + 255) / 256, 32), 256, 0, stream>>>(logits, ipl_out);
  class_logits_k<<<(32 * NCLS + 255) / 256, 256, 0, stream>>>(sa_w, ipl_out, cls_out);
  token_k<<<BN_TOK / 256, 256, 0, stream>>>(sam, labels, logits, pseudo, partmap, Aws, counts);
  accum_k<<<32, 768, 0, stream>>>(raw, rawinv, Aws, pseudo, labels, accbuf);
  ema_k<<<(CTOT * KPROT * DIMD + 255) / 256, 256, 0, stream>>>(proto, accbuf, counts, pnew);
}